// PhysicsInformedRounding_2396591751271
// MI455X (gfx1250) — compile-verified
//
#include <hip/hip_runtime.h>
#include <cstdint>

// Batched Kruskal MST edge selection for MI455X (gfx1250, wave32).
// One workgroup (256 threads = 8 wave32) per graph. Whole per-graph working
// set lives in LDS (CDNA5: 320KB/WGP). Probability tile staged via the
// Tensor Data Mover; the TDM transfer (TENSORcnt) is overlapped with the
// coalesced u/v global loads (LOADcnt) and union-find state init.

#define B_GRAPHS 256
#define E_EDGES  4096
#define N_NODES  2048
#define T_THREADS 256
#define EPT (E_EDGES / T_THREADS)   // 16 edges per thread

#if defined(__has_builtin)
#if __has_builtin(__builtin_amdgcn_tensor_load_to_lds) && \
    __has_builtin(__builtin_amdgcn_s_wait_tensorcnt)
#define USE_TDM 1
#endif
#endif

typedef __attribute__((ext_vector_type(4))) unsigned int v4u;
typedef __attribute__((ext_vector_type(8))) int          v8i;
typedef __attribute__((ext_vector_type(4))) int          v4i;

__global__ __launch_bounds__(T_THREADS)
void kruskal_select_kernel(const float* __restrict__ probs,
                           const int*   __restrict__ edge_index,
                           float*       __restrict__ out)
{
    // LDS working set: 32KB keys + 16KB packed uv + 8KB parent + 2KB rank + 512B mask
    __shared__ uint64_t s_key[E_EDGES];        // (~prob_bits)<<32 | orig_idx (sorted asc)
    __shared__ uint32_t s_uv[E_EDGES];         // staging for probs, then (u<<16)|v, co-sorted
    __shared__ int      s_parent[N_NODES];
    __shared__ uint8_t  s_rank[N_NODES];
    __shared__ uint32_t s_take[E_EDGES / 32];  // accept bitmask by ORIGINAL edge index

    const int g   = blockIdx.x;
    const int tid = threadIdx.x;
    const long long base = (long long)g * E_EDGES;
    const float* gp = probs + base;
    const int*   gu = edge_index + base;                                 // row 0: u
    const int*   gv = edge_index + (long long)B_GRAPHS * E_EDGES + base; // row 1: v

    // ---- Stage 1a: kick off async TDM copy of the 16KB probability tile ----
#ifdef USE_TDM
    if (tid == 0) {
        const uint64_t gaddr = (uint64_t)(uintptr_t)gp;
        const uint32_t laddr = (uint32_t)(uintptr_t)(&s_uv[0]); // LDS byte offset

        // D# group 0: count=1 | lds_addr | global_addr[56:0] | type=2
        v4u d0;
        d0.x = 1u;                                     // count=1, user mode
        d0.y = laddr;                                  // lds_addr
        d0.z = (uint32_t)gaddr;                        // global_addr[31:0]
        d0.w = (uint32_t)((gaddr >> 32) & 0x01FFFFFFu) // global_addr[56:32]
               | (2u << 30);                           // type=2 ("image")

        // D# group 1: data_size=4B, 1D: tensor_dim0=tile_dim0=4096, dim1=tile1=1
        v8i d1;
        d1[0] = (int)0x00020000u;                              // data_size=2 (4B)
        d1[1] = (int)(((uint32_t)E_EDGES & 0xFFFFu) << 16);    // tensor_dim0[15:0]
        d1[2] = (int)((1u << 16) | ((uint32_t)E_EDGES >> 16)); // dim1=1 | dim0[31:16]
        d1[3] = (int)((uint32_t)E_EDGES << 16);                // tile_dim0
        d1[4] = 1;                                             // tile_dim1=1
        d1[5] = E_EDGES;                                       // tensor_dim0_stride
        d1[6] = 0;
        d1[7] = 0;

        v4i dz = {0, 0, 0, 0};                                 // groups 2/3 unused (<=2D)
#if __clang_major__ >= 23
        v8i dz8 = {0, 0, 0, 0, 0, 0, 0, 0};
        __builtin_amdgcn_tensor_load_to_lds(d0, d1, dz, dz, dz8, 0);
#else
        __builtin_amdgcn_tensor_load_to_lds(d0, d1, dz, dz, 0);
#endif
        // NOTE: no wait here — TDM runs concurrently with the work below.
    }
#endif

    // ---- Stage 1b: overlapped with TDM: coalesced u/v loads + UF-state init ----
    uint32_t uvp[EPT];
    #pragma unroll
    for (int r = 0; r < EPT; ++r) {
        const int e = tid + r * T_THREADS;                // coalesced across the wave
        const uint32_t uu = (uint32_t)gu[e];
        const uint32_t vv = (uint32_t)gv[e];
        uvp[r] = (uu << 16) | (vv & 0xFFFFu);             // N=2048 fits in 11 bits
    }
    for (int n = tid; n < N_NODES; n += T_THREADS) { s_parent[n] = n; s_rank[n] = 0; }
    for (int w = tid; w < E_EDGES / 32; w += T_THREADS) s_take[w] = 0;

#ifdef USE_TDM
    if (tid == 0) __builtin_amdgcn_s_wait_tensorcnt(0);   // per-wave counter: wave 0 only
#else
    #pragma unroll
    for (int r = 0; r < EPT; ++r) {
        const int e = tid + r * T_THREADS;
        ((float*)s_uv)[e] = gp[e];
    }
#endif
    __syncthreads();

    // ---- Stage 2: build stable sort keys from staged probs; install packed uv ----
    float pr[EPT];
    #pragma unroll
    for (int r = 0; r < EPT; ++r) {
        const int e = tid + r * T_THREADS;
        const float p = ((const float*)s_uv)[e];  // staged probs (conflict-free stride)
        pr[r] = p;
        const uint32_t pb = __float_as_uint(p);   // p in [0,1): bits order-monotonic
        s_key[e] = ((uint64_t)(~pb) << 32) | (uint32_t)e; // asc sort => desc prob, stable
        s_uv[e] = uvp[r];                         // same thread, same slot: no hazard
    }
    __syncthreads();

    // ---- Stage 3: bitonic sort of (key, uv) pairs in LDS, ascending by key ----
    for (int k = 2; k <= E_EDGES; k <<= 1) {
        for (int j = k >> 1; j > 0; j >>= 1) {
            #pragma unroll
            for (int r = 0; r < EPT; ++r) {
                const int i   = tid + r * T_THREADS;
                const int ixj = i ^ j;
                if (ixj > i) {
                    const bool up = ((i & k) == 0);
                    const uint64_t a = s_key[i];
                    const uint64_t b = s_key[ixj];
                    if ((a > b) == up) {
                        s_key[i] = b; s_key[ixj] = a;
                        const uint32_t t = s_uv[i]; s_uv[i] = s_uv[ixj]; s_uv[ixj] = t;
                    }
                }
            }
            __syncthreads();
        }
    }

    // ---- Stage 4: serial union-find over sorted edges (thread 0) ----
    // The reference's cnt<target gate is provably never binding (accepted <=
    // actual_nodes-1 = target), and accept decisions depend only on components,
    // so path-halving is exact. Rank tie-break matches reference (tie: rv under ru).
    if (tid == 0) {
        for (int s = 0; s < E_EDGES; ++s) {
            const uint32_t uv = s_uv[s];
            int ru = (int)(uv >> 16);
            int rv = (int)(uv & 0xFFFFu);
            while (s_parent[ru] != ru) { const int pp = s_parent[s_parent[ru]]; s_parent[ru] = pp; ru = pp; }
            while (s_parent[rv] != rv) { const int pp = s_parent[s_parent[rv]]; s_parent[rv] = pp; rv = pp; }
            if (ru != rv) {
                const int rku = s_rank[ru], rkv = s_rank[rv];
                const int child = (rku < rkv) ? ru : rv;
                const int root  = (rku < rkv) ? rv : ru;
                s_parent[child] = root;
                if (rku == rkv) s_rank[root] = (uint8_t)(rku + 1);
                const uint32_t idx = (uint32_t)(s_key[s] & 0xFFFFFFFFull);
                s_take[idx >> 5] |= (1u << (idx & 31u));
            }
        }
    }
    __syncthreads();

    // ---- Stage 5: emit (dec - p) + p in the reference's exact fp32 op order ----
    #pragma unroll
    for (int r = 0; r < EPT; ++r) {
        const int e = tid + r * T_THREADS;
        const float d = (float)((s_take[e >> 5] >> (e & 31u)) & 1u);
        const float p = pr[r];
        out[base + e] = (d - p) + p;
    }
}

extern "C" void kernel_launch(void* const* d_in, const int* in_sizes, int n_in,
                              void* d_out, int out_size, void* d_ws, size_t ws_size,
                              hipStream_t stream) {
    (void)in_sizes; (void)n_in; (void)out_size; (void)d_ws; (void)ws_size;
    const float* probs      = (const float*)d_in[0]; // switch_probs [B*E] f32
    const int*   edge_index = (const int*)d_in[1];   // edge_index [2, B*E] i32
    // d_in[2] edge_batch and d_in[3] num_nodes_per_graph are not needed:
    // edges are contiguous per graph and the target-count gate is dead code.
    float* out = (float*)d_out;                      // decisions [B*E] f32
    kruskal_select_kernel<<<B_GRAPHS, T_THREADS, 0, stream>>>(probs, edge_index, out);
}